// TAMCaD_13778255085577
// MI455X (gfx1250) — compile-verified
//
#include <hip/hip_runtime.h>
#include <math.h>

typedef float v2f __attribute__((ext_vector_type(2)));
typedef float v4f __attribute__((ext_vector_type(4)));
typedef float v8f __attribute__((ext_vector_type(8)));

#define B_   4
#define G_   64
#define H_   4
#define T_   1024
#define TPB  4      // t values per workgroup
#define NCH  512    // G_*DPG_ channels

// LDS layout (float offsets)
#define SQ_OFF 0                          // [512][4]
#define SK_OFF (SQ_OFF + NCH * TPB)       // [512][5]
#define SV_OFF (SK_OFF + NCH * (TPB+1))   // [512][5]
#define SX_OFF (SV_OFF + NCH * (TPB+1))   // [16 waveslots][64 rows][16 j]
#define SH_OFF (SX_OFF + 16 * 64 * 16)    // [16 waveslots][64 g]
#define LDS_FLOATS (SH_OFF + 16 * 64)     // 24576 floats = 96 KB

__global__ __launch_bounds__(512)
void tamcad_attn_kernel(const float* __restrict__ qkv,
                        const float* __restrict__ w1,
                        const float* __restrict__ b1,
                        float* __restrict__ pred,
                        float* __restrict__ attn)
{
    extern __shared__ float lds[];
    float* sQ = lds + SQ_OFF;
    float* sK = lds + SK_OFF;
    float* sV = lds + SV_OFF;
    float* sX = lds + SX_OFF;
    float* sH = lds + SH_OFF;

    const int tid = threadIdx.x;
    const int blk = blockIdx.x;
    const int b   = blk >> 8;            // 256 t-blocks per batch
    const int t0  = (blk & 255) * TPB;

    // ---- stage qkv slices into LDS (qkv: 100MB, resident in 192MB L2) ----
    const size_t baseBC = (size_t)b * 6144;
    for (int e = tid; e < NCH * TPB; e += 512) {
        int ch = e >> 2, tt = e & 3;
        sQ[e] = qkv[(baseBC + (size_t)ch * 3 + 0) * 1025 + t0 + tt];
    }
    for (int e = tid; e < NCH * (TPB + 1); e += 512) {
        int ch = e / 5, tt = e - ch * 5;
        sK[e] = qkv[(baseBC + (size_t)ch * 3 + 1) * 1025 + t0 + tt];
    }
    for (int e = tid; e < NCH * (TPB + 1); e += 512) {
        int ch = e / 5, tt = e - ch * 5;
        sV[e] = qkv[(baseBC + (size_t)ch * 3 + 2) * 1025 + t0 + tt];
    }
    __syncthreads();

    const int wv = tid >> 5;         // wave slot 0..15
    const int h  = wv >> 2;          // head 0..3
    const int tl = wv & 3;           // local t 0..3
    const int ln = tid & 31;
    const int lh = ln >> 4;          // half-wave
    const int lm = ln & 15;
    const int xbase = wv * 1024;     // this wave's 64x16 slab in sX

    // ---- Q A-fragments (A layout: lane = M, k-pair per half), scale folded ----
    const float scale = 0.35355339059327373f;  // 1/sqrt(8)
    v2f aq[4][2];
    #pragma unroll
    for (int mb = 0; mb < 4; ++mb) {
        #pragma unroll
        for (int s = 0; s < 2; ++s) {
            int g = mb * 16 + lm;
            int d = s * 4 + lh * 2;
            aq[mb][s].x = sQ[(g * 32 + h * 8 + d    ) * TPB + tl] * scale;
            aq[mb][s].y = sQ[(g * 32 + h * 8 + d + 1) * TPB + tl] * scale;
        }
    }

    // lane-resident row stats: lane ln owns rows 2*ln and 2*ln+1
    float mA[2] = { -INFINITY, -INFINITY };
    float lA[2] = { 0.f, 0.f };

    // ---- sweep A: online row max + sum, reductions done in-lane via LDS slab ----
    for (int nb = 0; nb < 8; ++nb) {
        v2f bk[2];
        #pragma unroll
        for (int s = 0; s < 2; ++s) {
            int j = nb * 16 + lm;            // B layout: lane = N
            int jg = j & 63, sh = j >> 6;    // shifted K half uses t+1
            int d = s * 4 + lh * 2;
            bk[s].x = sK[(jg * 32 + h * 8 + d    ) * (TPB + 1) + tl + sh];
            bk[s].y = sK[(jg * 32 + h * 8 + d + 1) * (TPB + 1) + tl + sh];
        }
        #pragma unroll
        for (int mb = 0; mb < 4; ++mb) {
            v8f c = {0.f,0.f,0.f,0.f,0.f,0.f,0.f,0.f};
            c = __builtin_amdgcn_wmma_f32_16x16x4_f32(false, aq[mb][0], false, bk[0], (short)0, c, false, false);
            c = __builtin_amdgcn_wmma_f32_16x16x4_f32(false, aq[mb][1], false, bk[1], (short)0, c, false, false);
            #pragma unroll
            for (int r = 0; r < 8; ++r)      // C layout -> row-major slab
                sX[xbase + (16 * mb + r + 8 * lh) * 16 + lm] = c[r];
        }
        // each lane reduces its 2 rows (same-wave LDS is hardware in-order)
        #pragma unroll
        for (int rr = 0; rr < 2; ++rr) {
            const float* rowp = sX + xbase + (2 * ln + rr) * 16;
            v4f x0 = *(const v4f*)(rowp + 0);
            v4f x1 = *(const v4f*)(rowp + 4);
            v4f x2 = *(const v4f*)(rowp + 8);
            v4f x3 = *(const v4f*)(rowp + 12);
            v4f mx4 = __builtin_elementwise_max(__builtin_elementwise_max(x0, x1),
                                                __builtin_elementwise_max(x2, x3));
            float bm = fmaxf(fmaxf(mx4.x, mx4.y), fmaxf(mx4.z, mx4.w));
            float nm = fmaxf(mA[rr], bm);
            float sum = 0.f;
            #pragma unroll
            for (int q = 0; q < 4; ++q) {
                sum += __expf(x0[q] - nm); sum += __expf(x1[q] - nm);
                sum += __expf(x2[q] - nm); sum += __expf(x3[q] - nm);
            }
            lA[rr] = lA[rr] * __expf(mA[rr] - nm) + sum;
            mA[rr] = nm;
        }
    }
    float rlA[2] = { 1.0f / lA[0], 1.0f / lA[1] };

    // ---- sweep B: recompute, normalize in slab, stream attn, accumulate P·V ----
    v8f o[4];
    #pragma unroll
    for (int mb = 0; mb < 4; ++mb) o[mb] = (v8f){0.f,0.f,0.f,0.f,0.f,0.f,0.f,0.f};

    for (int nb = 0; nb < 8; ++nb) {
        v2f bk[2];
        #pragma unroll
        for (int s = 0; s < 2; ++s) {
            int j = nb * 16 + lm;
            int jg = j & 63, sh = j >> 6;
            int d = s * 4 + lh * 2;
            bk[s].x = sK[(jg * 32 + h * 8 + d    ) * (TPB + 1) + tl + sh];
            bk[s].y = sK[(jg * 32 + h * 8 + d + 1) * (TPB + 1) + tl + sh];
        }
        #pragma unroll
        for (int mb = 0; mb < 4; ++mb) {
            v8f c = {0.f,0.f,0.f,0.f,0.f,0.f,0.f,0.f};
            c = __builtin_amdgcn_wmma_f32_16x16x4_f32(false, aq[mb][0], false, bk[0], (short)0, c, false, false);
            c = __builtin_amdgcn_wmma_f32_16x16x4_f32(false, aq[mb][1], false, bk[1], (short)0, c, false, false);
            #pragma unroll
            for (int r = 0; r < 8; ++r)
                sX[xbase + (16 * mb + r + 8 * lh) * 16 + lm] = c[r];
        }
        // in-lane softmax normalization of this lane's 2 rows, in place
        #pragma unroll
        for (int rr = 0; rr < 2; ++rr) {
            float* rowp = sX + xbase + (2 * ln + rr) * 16;
            #pragma unroll
            for (int qb = 0; qb < 4; ++qb) {
                v4f x = *(const v4f*)(rowp + 4 * qb);
                v4f p;
                p.x = __expf(x.x - mA[rr]) * rlA[rr];
                p.y = __expf(x.y - mA[rr]) * rlA[rr];
                p.z = __expf(x.z - mA[rr]) * rlA[rr];
                p.w = __expf(x.w - mA[rr]) * rlA[rr];
                *(v4f*)(rowp + 4 * qb) = p;
            }
        }
        __syncthreads();

        // cooperative t-contiguous NT stores of the workgroup's attn block
        for (int rr2 = tid; rr2 < H_ * 64 * 16; rr2 += 512) {
            int h2 = rr2 >> 10, rem = rr2 & 1023;
            int row2 = rem >> 4, lm2 = rem & 15;
            v4f v4;
            v4.x = sX[(h2 * 4 + 0) * 1024 + row2 * 16 + lm2];
            v4.y = sX[(h2 * 4 + 1) * 1024 + row2 * 16 + lm2];
            v4.z = sX[(h2 * 4 + 2) * 1024 + row2 * 16 + lm2];
            v4.w = sX[(h2 * 4 + 3) * 1024 + row2 * 16 + lm2];
            size_t off = ((((size_t)b * H_ + h2) * 64 + row2) * 128 + nb * 16 + lm2) * (size_t)T_ + t0;
            __builtin_nontemporal_store(v4, (v4f*)(attn + off));
        }

        // P·V accumulate: A-frags are contiguous b64 loads from own slab
        #pragma unroll
        for (int s2 = 0; s2 < 4; ++s2) {
            int jl = s2 * 4 + lh * 2;
            int j0 = nb * 16 + jl, j1 = j0 + 1;
            v2f bv;
            bv.x = (lm < 8) ? sV[((j0 & 63) * 32 + h * 8 + lm) * (TPB + 1) + tl + (j0 >> 6)] : 0.f;
            bv.y = (lm < 8) ? sV[((j1 & 63) * 32 + h * 8 + lm) * (TPB + 1) + tl + (j1 >> 6)] : 0.f;
            #pragma unroll
            for (int mb = 0; mb < 4; ++mb) {
                v2f ap = *(const v2f*)(sX + xbase + (16 * mb + lm) * 16 + jl);
                o[mb] = __builtin_amdgcn_wmma_f32_16x16x4_f32(false, ap, false, bv, (short)0, o[mb], false, false);
            }
        }
        __syncthreads();
    }

    // ---- pred: dump O to slab, per-lane row dot with fc1_w, cross-head LDS sum ----
    #pragma unroll
    for (int mb = 0; mb < 4; ++mb)
        #pragma unroll
        for (int r = 0; r < 8; ++r)
            sX[xbase + (16 * mb + r + 8 * lh) * 16 + lm] = o[mb][r];
    #pragma unroll
    for (int rr = 0; rr < 2; ++rr) {
        int g2 = 2 * ln + rr;
        const float* op = sX + xbase + g2 * 16;       // O row: dk 0..7 (8..15 are zero)
        v4f o0 = *(const v4f*)(op + 0);
        v4f o1 = *(const v4f*)(op + 4);
        v4f w0 = *(const v4f*)(w1 + g2 * 32 + h * 8 + 0);
        v4f w4 = *(const v4f*)(w1 + g2 * 32 + h * 8 + 4);
        v4f pr = o0 * w0 + o1 * w4;
        sH[wv * 64 + g2] = (pr.x + pr.y) + (pr.z + pr.w);
    }
    __syncthreads();
    if (tid < 256) {
        int tl3 = tid >> 6, g3 = tid & 63;
        float acc = b1[g3];
        #pragma unroll
        for (int hh = 0; hh < H_; ++hh) acc += sH[(hh * 4 + tl3) * 64 + g3];
        pred[(size_t)b * (G_ * T_) + (size_t)g3 * T_ + t0 + tl3] = acc;
    }
}

extern "C" void kernel_launch(void* const* d_in, const int* in_sizes, int n_in,
                              void* d_out, int out_size, void* d_ws, size_t ws_size,
                              hipStream_t stream) {
    const float* qkv = (const float*)d_in[0];
    const float* w1  = (const float*)d_in[1];
    const float* b1  = (const float*)d_in[2];
    float* pred = (float*)d_out;                          // B*G*T floats
    float* attn = (float*)d_out + (size_t)B_ * G_ * T_;   // B*H*G*2G*T floats
    dim3 grid(B_ * (T_ / TPB));   // 1024 workgroups
    size_t smem = (size_t)LDS_FLOATS * sizeof(float);     // 96 KB dynamic LDS
    tamcad_attn_kernel<<<grid, 512, smem, stream>>>(qkv, w1, b1, pred, attn);
}